// Gaussian3D_67113158967483
// MI455X (gfx1250) — compile-verified
//
#include <hip/hip_runtime.h>
#include <math.h>

#define N_GAUSS 4096
#define IMG_H 384
#define IMG_W 384
#define FOCAL_C 192.0f
#define CX_C 192.0f
#define CY_C 192.0f
#define CLIPZ_C 0.01f
#define BLUR_C 0.3f
#define AMIN_C (1.0f/255.0f)
#define GB 64
#define COLP (GB + 2)   // row pitch (dwords): distinct banks per row, 8B aligned

typedef __attribute__((ext_vector_type(2))) float v2f;
typedef __attribute__((ext_vector_type(8))) float v8f;

__device__ __forceinline__ float sigmoidf(float v) { return 1.0f / (1.0f + __expf(-v)); }

// ---------------- Kernel 1: per-Gaussian projection / conic ----------------
__global__ __launch_bounds__(256) void gs_preprocess(
    const float* __restrict__ xyz, const float* __restrict__ scaling,
    const float* __restrict__ opacity, const float* __restrict__ rotation,
    const float* __restrict__ fdc,
    float* __restrict__ key,
    float* __restrict__ PX, float* __restrict__ PY,
    float* __restrict__ CA, float* __restrict__ CB, float* __restrict__ CC,
    float* __restrict__ OP, float* __restrict__ CR, float* __restrict__ CG,
    float* __restrict__ CBl)
{
  int i = blockIdx.x * blockDim.x + threadIdx.x;
  if (i >= N_GAUSS) return;

  // view transform: Rcam = I, t = (0,0,8)
  float x = xyz[3*i+0], y = xyz[3*i+1], z = xyz[3*i+2] + 8.0f;
  float zs = (z > CLIPZ_C) ? z : 1.0f;

  float s0 = __expf(scaling[3*i+0]);
  float s1 = __expf(scaling[3*i+1]);
  float s2 = __expf(scaling[3*i+2]);

  float qw = rotation[4*i+0], qx = rotation[4*i+1],
        qy = rotation[4*i+2], qz = rotation[4*i+3];
  float qn = 1.0f / sqrtf(qw*qw + qx*qx + qy*qy + qz*qz);
  qw *= qn; qx *= qn; qy *= qn; qz *= qn;

  // M = Rq * diag(s)  (column scaled)
  float m00 = (1.0f - 2.0f*(qy*qy + qz*qz)) * s0;
  float m01 = (2.0f*(qx*qy - qw*qz)) * s1;
  float m02 = (2.0f*(qx*qz + qw*qy)) * s2;
  float m10 = (2.0f*(qx*qy + qw*qz)) * s0;
  float m11 = (1.0f - 2.0f*(qx*qx + qz*qz)) * s1;
  float m12 = (2.0f*(qy*qz - qw*qx)) * s2;
  float m20 = (2.0f*(qx*qz - qw*qy)) * s0;
  float m21 = (2.0f*(qy*qz + qw*qx)) * s1;
  float m22 = (1.0f - 2.0f*(qx*qx + qy*qy)) * s2;

  // cov3d = M M^T (symmetric entries we need)
  float C00 = m00*m00 + m01*m01 + m02*m02;
  float C01 = m00*m10 + m01*m11 + m02*m12;
  float C02 = m00*m20 + m01*m21 + m02*m22;
  float C11 = m10*m10 + m11*m11 + m12*m12;
  float C12 = m10*m20 + m11*m21 + m12*m22;
  float C22 = m20*m20 + m21*m21 + m22*m22;

  float rz  = 1.0f / zs;
  float rz2 = rz * rz;
  // tan_fovx = tan_fovy = 1.0
  float txc = zs * fminf(fmaxf(x * rz, -1.3f), 1.3f);
  float tyc = zs * fminf(fmaxf(y * rz, -1.3f), 1.3f);
  float fz  = FOCAL_C * rz;
  float jx  = -FOCAL_C * txc * rz2;
  float jy  = -FOCAL_C * tyc * rz2;

  // cov2d = J * cov3d * J^T (Rcam = I so T = J)
  float c00 = fz*fz*C00 + 2.0f*fz*jx*C02 + jx*jx*C22 + BLUR_C;
  float c01 = fz*fz*C01 + fz*jy*C02 + fz*jx*C12 + jx*jy*C22;
  float c11 = fz*fz*C11 + 2.0f*fz*jy*C12 + jy*jy*C22 + BLUR_C;

  float det = c00*c11 - c01*c01;
  bool valid = (z > CLIPZ_C) && (det > 0.0f);
  float dinv = 1.0f / (valid ? det : 1.0f);

  CA[i] =  c11 * dinv;
  CB[i] = -c01 * dinv;
  CC[i] =  c00 * dinv;
  PX[i] = FOCAL_C * x * rz + CX_C;
  PY[i] = FOCAL_C * y * rz + CY_C;
  CR[i]  = sigmoidf(fdc[3*i+0]);
  CG[i]  = sigmoidf(fdc[3*i+1]);
  CBl[i] = sigmoidf(fdc[3*i+2]);
  OP[i]  = sigmoidf(opacity[i]) * (valid ? 1.0f : 0.0f);
  key[i] = valid ? z : __builtin_inff();
}

// ---------------- Kernel 2: single-WGP LDS bitonic argsort ----------------
__global__ __launch_bounds__(1024) void gs_sort(const float* __restrict__ key,
                                                int* __restrict__ idx)
{
  __shared__ float sk[N_GAUSS];
  __shared__ int   sv[N_GAUSS];
  int tid = threadIdx.x;
  for (int i = tid; i < N_GAUSS; i += 1024) { sk[i] = key[i]; sv[i] = i; }
  __syncthreads();

  for (int size = 2; size <= N_GAUSS; size <<= 1) {
    for (int stride = size >> 1; stride > 0; stride >>= 1) {
      for (int t = tid; t < N_GAUSS / 2; t += 1024) {
        int i = ((t & ~(stride - 1)) << 1) | (t & (stride - 1));
        int j = i | stride;
        bool up = ((i & size) == 0);
        float ki = sk[i], kj = sk[j];
        if ((ki > kj) == up) {
          sk[i] = kj; sk[j] = ki;
          int vi = sv[i]; sv[i] = sv[j]; sv[j] = vi;
        }
      }
      __syncthreads();
    }
  }
  for (int i = tid; i < N_GAUSS; i += 1024) idx[i] = sv[i];
}

// ---------------- Kernel 3: tiled rasterizer, WMMA f32 color accumulate ----
// Tile = 16x8 pixels. 8 waves/WG; wave w owns row y=w of the tile (16 pixels).
// Per wave and 4-splat chunk:
//   A(16x4 f32)  = compositing weights  (lanes 0-15: K=0,1; lanes 16-31: K=2,3)
//   B(4x16 f32)  = splat RGB^T, columns 0..2 live, 3..15 zero
//                  (lanes 0-15: rows K=0,1; lanes 16-31: rows K=2,3)
//   C/D(16x16)   = running image accumulator (8 VGPRs, exact f32)
// B is fetched branchlessly from a zero-padded 16-row LDS color table, one
// ds_load_b64 per lane; rows are pitch-66 so the 16 rows map to 16 banks.
__global__ __launch_bounds__(256) void gs_raster(
    const int* __restrict__ order,
    const float* __restrict__ PX, const float* __restrict__ PY,
    const float* __restrict__ CA, const float* __restrict__ CB,
    const float* __restrict__ CC, const float* __restrict__ OP,
    const float* __restrict__ CR, const float* __restrict__ CG,
    const float* __restrict__ CBl, float* __restrict__ out)
{
  __shared__ float sgx[GB], sgy[GB], sca[GB], scb[GB], scc[GB], sop[GB];
  __shared__ float scol[16][COLP];    // B-matrix source: rows 0..2 rgb, rest 0
  __shared__ float dsc[8 * 32 * 8];   // per-wave D spill for final extraction

  int tid  = threadIdx.x;
  int wave = tid >> 5;
  int lane = tid & 31;
  int hf   = lane >> 4;       // which K-half this lane supplies
  int m    = lane & 15;       // matrix row / B column == pixel x within row

  // zero the color table once (rows 3..15 stay zero forever)
  for (int i = tid; i < 16 * COLP; i += 256) (&scol[0][0])[i] = 0.0f;

  int px = blockIdx.x * 16 + m;
  int py = blockIdx.y * 8 + wave;
  float fx = (float)px, fy = (float)py;

  const float* mycol = &scol[m][0];   // per-lane B-row base (branchless)

  v8f acc = (v8f)0.0f;        // D/C accumulator (16x16 f32)
  float T = 1.0f;             // per-pixel transmittance (dup on lane pair)

  for (int base = 0; base < N_GAUSS; base += GB) {
    __syncthreads();
    if (tid < GB) {
      int gi = order[base + tid];
      sgx[tid] = PX[gi];  sgy[tid] = PY[gi];
      sca[tid] = CA[gi];  scb[tid] = CB[gi];  scc[tid] = CC[gi];
      sop[tid] = OP[gi];
      scol[0][tid] = CR[gi];
      scol[1][tid] = CG[gi];
      scol[2][tid] = CBl[gi];
    }
    if (base + GB < N_GAUSS)  // hide next block's index gather (global_prefetch_b8)
      __builtin_prefetch(&order[base + GB + (tid & 31)], 0, 1);
    __syncthreads();

    for (int c4 = 0; c4 < GB; c4 += 4) {
      float w[4];
      #pragma unroll
      for (int j = 0; j < 4; ++j) {
        int g = c4 + j;
        float dx  = sgx[g] - fx;
        float dy  = sgy[g] - fy;
        float sig = 0.5f * (sca[g]*dx*dx + scc[g]*dy*dy) + scb[g]*dx*dy;
        float a   = fminf(0.999f, sop[g] * __expf(-sig));
        a = (sig >= 0.0f && a >= AMIN_C) ? a : 0.0f;
        w[j] = T * a;          // Tpre * alpha
        T *= (1.0f - a);       // serial front-to-back transmittance
      }
      int k0 = c4 + (hf << 1);
      v2f A;
      A.x = hf ? w[2] : w[0];
      A.y = hf ? w[3] : w[1];
      v2f Bv;                  // one ds_load_b64, no divergence
      Bv.x = mycol[k0];
      Bv.y = mycol[k0 + 1];

      acc = __builtin_amdgcn_wmma_f32_16x16x4_f32(
          false, A, false, Bv, (short)0, acc, false, false);
    }
  }

  // Extract D: element D[M][N] lives in lane (N + 16*(M>=8)), VGPR M%8.
  __syncthreads();
  float* ds = &dsc[wave * 256];
  #pragma unroll
  for (int r = 0; r < 8; ++r) ds[lane * 8 + r] = acc[r];
  __syncthreads();

  if (hf == 0) {
    int hi = (m >= 8) ? 16 : 0;
    #pragma unroll
    for (int c = 0; c < 3; ++c) {
      float v = ds[(c + hi) * 8 + (m & 7)] + T;   // + white background * T
      v = fminf(v, 1.0f);
      out[c * IMG_H * IMG_W + py * IMG_W + px] = v;
    }
  }
}

// ---------------- Host-side launch ----------------
extern "C" void kernel_launch(void* const* d_in, const int* in_sizes, int n_in,
                              void* d_out, int out_size, void* d_ws, size_t ws_size,
                              hipStream_t stream)
{
  (void)in_sizes; (void)n_in; (void)out_size; (void)ws_size;
  const float* xyz      = (const float*)d_in[0];
  const float* scaling  = (const float*)d_in[1];
  const float* opacity  = (const float*)d_in[2];
  const float* rotation = (const float*)d_in[3];
  const float* fdc      = (const float*)d_in[4];
  float* out = (float*)d_out;

  float* ws   = (float*)d_ws;
  float* key  = ws + 0 * N_GAUSS;
  int*   idx  = (int*)(ws + 1 * N_GAUSS);
  float* PX   = ws + 2 * N_GAUSS;
  float* PY   = ws + 3 * N_GAUSS;
  float* CAa  = ws + 4 * N_GAUSS;
  float* CBa  = ws + 5 * N_GAUSS;
  float* CCa  = ws + 6 * N_GAUSS;
  float* OPa  = ws + 7 * N_GAUSS;
  float* CRa  = ws + 8 * N_GAUSS;
  float* CGa  = ws + 9 * N_GAUSS;
  float* CBl  = ws + 10 * N_GAUSS;

  gs_preprocess<<<N_GAUSS / 256, 256, 0, stream>>>(
      xyz, scaling, opacity, rotation, fdc,
      key, PX, PY, CAa, CBa, CCa, OPa, CRa, CGa, CBl);

  gs_sort<<<1, 1024, 0, stream>>>(key, idx);

  gs_raster<<<dim3(IMG_W / 16, IMG_H / 8), 256, 0, stream>>>(
      idx, PX, PY, CAa, CBa, CCa, OPa, CRa, CGa, CBl, out);
}